// RecurrentGCN_60997125538332
// MI455X (gfx1250) — compile-verified
//
#include <hip/hip_runtime.h>
#include <hip/hip_bf16.h>

#define NNODES 50000
#define NEDGES 800000
#define CH     64
#define NC     (NNODES * CH)
#define KORD   3
#define BN_EPS 1e-5f

typedef __attribute__((ext_vector_type(2))) float v2f;
typedef __attribute__((ext_vector_type(8))) float v8f;

__device__ __forceinline__ float sigm(float v) { return 1.0f / (1.0f + expf(-v)); }
__device__ __forceinline__ float bnrm(float v, float g, float b, float m, float var) {
    return (v - m) * rsqrtf(var + BN_EPS) * g + b;
}

// ---------------- elementwise / graph kernels ----------------

__global__ void k_zero(float* __restrict__ p, int n) {
    int i = blockIdx.x * blockDim.x + threadIdx.x;
    if (i < n) p[i] = 0.0f;
}

__global__ void k_scalecopy(float* __restrict__ y, const float* __restrict__ x, float a, int n) {
    int i = blockIdx.x * blockDim.x + threadIdx.x;
    if (i < n) y[i] = a * x[i];
}

// deg[src[e]] += w[e]   (one thread per edge)
__global__ void k_deg(const int* __restrict__ src, const float* __restrict__ w,
                      float* __restrict__ deg, int E) {
    int e = blockIdx.x * blockDim.x + threadIdx.x;
    if (e < E) unsafeAtomicAdd(&deg[src[e]], w[e]);
}

// in-place: deg -> (deg>0 ? rsqrt(deg) : 0)
__global__ void k_dis(float* __restrict__ d, int n) {
    int i = blockIdx.x * blockDim.x + threadIdx.x;
    if (i < n) { float v = d[i]; d[i] = (v > 0.0f) ? rsqrtf(v) : 0.0f; }
}

// norm[e] = -dis[src]*w*dis[dst]
__global__ void k_norm(const int* __restrict__ src, const int* __restrict__ dst,
                       const float* __restrict__ w, const float* __restrict__ dis,
                       float* __restrict__ norm, int E) {
    int e = blockIdx.x * blockDim.x + threadIdx.x;
    if (e < E) norm[e] = -dis[src[e]] * w[e] * dis[dst[e]];
}

// y[dst] += scale*norm[e]*x[src]; one wave per edge, 2 channels per lane
__global__ void k_scatter(const float* __restrict__ x, const float* __restrict__ norm,
                          const int* __restrict__ src, const int* __restrict__ dst,
                          float* __restrict__ y, float scale, int E) {
    long tid = (long)blockIdx.x * blockDim.x + threadIdx.x;
    int e = (int)(tid >> 5);
    if (e >= E) return;
    int lane = (int)(tid & 31);
    int s = src[e], d = dst[e];
    float w = norm[e] * scale;
    const v2f v = *(const v2f*)(x + (long)s * CH + lane * 2);
    float* yp = y + (long)d * CH + lane * 2;
    unsafeAtomicAdd(yp,     w * v[0]);
    unsafeAtomicAdd(yp + 1, w * v[1]);
}

// layer-1 combine: h1 = relu(bn1((1 - sigmoid(Gz)) * tanh(Gh)))
__global__ void k_gate1(const float* __restrict__ Gz, const float* __restrict__ Gh,
                        const float* __restrict__ g, const float* __restrict__ b,
                        const float* __restrict__ m, const float* __restrict__ v,
                        float* __restrict__ h1, int n) {
    int i = blockIdx.x * blockDim.x + threadIdx.x;
    if (i >= n) return;
    int c = i & (CH - 1);
    float z = sigm(Gz[i]);
    float ht = tanhf(Gh[i]);
    float val = bnrm((1.0f - z) * ht, g[c], b[c], m[c], v[c]);
    h1[i] = fmaxf(val, 0.0f);
}

// HR = H * sigmoid(Rpre)   (in-place over Rpre is fine)
__global__ void k_hr(const float* __restrict__ Rpre, const float* __restrict__ H,
                     float* __restrict__ hr, int n) {
    int i = blockIdx.x * blockDim.x + threadIdx.x;
    if (i < n) hr[i] = H[i] * sigm(Rpre[i]);
}

// layer-2 combine: h = relu(bn2(z*H + (1-z)*tanh(Htpre)))
__global__ void k_gate2(const float* __restrict__ Zpre, const float* __restrict__ Htpre,
                        const float* __restrict__ H,
                        const float* __restrict__ g, const float* __restrict__ b,
                        const float* __restrict__ m, const float* __restrict__ v,
                        float* __restrict__ hout, int n) {
    int i = blockIdx.x * blockDim.x + threadIdx.x;
    if (i >= n) return;
    int c = i & (CH - 1);
    float z = sigm(Zpre[i]);
    float ht = tanhf(Htpre[i]);
    float val = bnrm(z * H[i] + (1.0f - z) * ht, g[c], b[c], m[c], v[c]);
    hout[i] = fmaxf(val, 0.0f);
}

// ---------------- WMMA GEMM: out = [accum +] sum_t Xt @ Wt + bias0 + bias1 ----------------
// One wave -> 16x16 f32 tile via V_WMMA_F32_16X16X4_F32; 4 waves/block cover 64 cols.
template <int NTERMS, bool TRANSB>
__global__ __launch_bounds__(128)
void k_mm_wmma(const float* __restrict__ X0, const float* __restrict__ X1,
               const float* __restrict__ X2, const float* __restrict__ W,
               const float* __restrict__ bias0, const float* __restrict__ bias1,
               float* __restrict__ out, int accumulate) {
    const int lane = threadIdx.x & 31;
    const int wv   = threadIdx.x >> 5;     // column tile 0..3
    const int row0 = blockIdx.x * 16;
    const int n0   = wv * 16;
    const int l15  = lane & 15;
    const int hi   = lane >> 4;            // 0 or 1

    v8f c = {0.f, 0.f, 0.f, 0.f, 0.f, 0.f, 0.f, 0.f};

    const float* Xs[3] = {X0, X1, X2};
#pragma unroll
    for (int t = 0; t < NTERMS; ++t) {
        const float* __restrict__ X  = Xs[t];
        const float* __restrict__ Wt = W + t * CH * CH;
#pragma unroll
        for (int kk = 0; kk < CH; kk += 4) {
            const int ak = kk + hi * 2;
            // A fragment: 16x4, lane<16 -> (M=l15, K=kk..kk+1); lane>=16 -> K=kk+2..kk+3
            v2f a = *(const v2f*)(X + (long)(row0 + l15) * CH + ak);
            // B fragment: 4x16, (K=ak, N=n0+l15) and (K=ak+1, N=n0+l15)
            v2f b;
            if (TRANSB) {
                const float* p = Wt + (long)(n0 + l15) * CH + ak;
                b[0] = p[0]; b[1] = p[1];
            } else {
                const float* p = Wt + (long)ak * CH + (n0 + l15);
                b[0] = p[0]; b[1] = p[CH];
            }
            c = __builtin_amdgcn_wmma_f32_16x16x4_f32(false, a, false, b,
                                                      (short)0, c, false, false);
        }
    }

    const int ncol = n0 + l15;
    float bv = 0.0f;
    if (bias0) bv += bias0[ncol];
    if (bias1) bv += bias1[ncol];

    // C/D layout: VGPR r -> M = r + (hi ? 8 : 0), N = ncol
    const int m0 = row0 + hi * 8;
    float* o = out + (long)m0 * CH + ncol;
#pragma unroll
    for (int r = 0; r < 8; ++r) {
        float val = c[r] + bv;
        if (accumulate) val += o[(long)r * CH];
        o[(long)r * CH] = val;
    }
}

// ---------------- host orchestration ----------------

extern "C" void kernel_launch(void* const* d_in, const int* in_sizes, int n_in,
                              void* d_out, int out_size, void* d_ws, size_t ws_size,
                              hipStream_t stream) {
    (void)in_sizes; (void)n_in; (void)out_size; (void)ws_size;
    const int N = NNODES, E = NEDGES;

    const float* x     = (const float*)d_in[0];
    const int*   ei    = (const int*)d_in[1];
    const int*   src   = ei;
    const int*   dst   = ei + E;
    const float* ew    = (const float*)d_in[2];
    const float* H     = (const float*)d_in[3];
    const float* l1_Wx = (const float*)d_in[4];
    const float* l1_Wh = (const float*)d_in[5];   // unused except biases (H=0)
    const float* l1_bx = (const float*)d_in[6];
    const float* l1_bh = (const float*)d_in[7];
    const float* l2_Wx = (const float*)d_in[8];
    const float* l2_Wh = (const float*)d_in[9];
    const float* l2_bx = (const float*)d_in[10];
    const float* l2_bh = (const float*)d_in[11];
    const float* bn1_g = (const float*)d_in[12];
    const float* bn1_b = (const float*)d_in[13];
    const float* bn1_m = (const float*)d_in[14];
    const float* bn1_v = (const float*)d_in[15];
    const float* bn2_g = (const float*)d_in[16];
    const float* bn2_b = (const float*)d_in[17];
    const float* bn2_m = (const float*)d_in[18];
    const float* bn2_v = (const float*)d_in[19];
    const float* headW = (const float*)d_in[20];
    const float* headb = (const float*)d_in[21];
    (void)l1_Wh;

    float* ws   = (float*)d_ws;
    float* norm = ws;                 // [E]
    float* dis  = ws + E;             // [N]  (deg, then rsqrt in place)
    float* B0   = ws + E + N;         // 8 buffers of [N*CH]
    float* B1   = B0 + NC;
    float* B2   = B1 + NC;
    float* B3   = B2 + NC;
    float* B4   = B3 + NC;            // h1
    float* B5   = B4 + NC;            // Z pre-activation
    float* B6   = B5 + NC;            // R pre-activation / HR
    float* B7   = B6 + NC;            // Ht pre-activation
    float* hout = (float*)d_out + NC; // second output: h
    float* oout = (float*)d_out;      // first output: head

    const int TB = 256;
    const int gE   = (E + TB - 1) / TB;          // edge-parallel
    const int gE32 = (E * 32 + TB - 1) / TB;     // wave-per-edge scatter
    const int gN   = (N + TB - 1) / TB;
    const int gNC  = (NC + TB - 1) / TB;
    const dim3 gMM(N / 16);                      // 3125 row tiles
    const dim3 bMM(128);                         // 4 waves (wave32)

    const int KCC = KORD * CH * CH;

    // ---- normalized edge weights ----
    k_zero<<<gN, TB, 0, stream>>>(dis, N);
    k_deg<<<gE, TB, 0, stream>>>(src, ew, dis, E);
    k_dis<<<gN, TB, 0, stream>>>(dis, N);
    k_norm<<<gE, TB, 0, stream>>>(src, dst, ew, dis, norm, E);

    // ---- layer 1 (hidden state = 0): T1 = prop(x), T2 = 2*prop(T1) - x ----
    k_zero<<<gNC, TB, 0, stream>>>(B0, NC);
    k_scatter<<<gE32, TB, 0, stream>>>(x, norm, src, dst, B0, 1.0f, E);
    k_scalecopy<<<gNC, TB, 0, stream>>>(B1, x, -1.0f, NC);
    k_scatter<<<gE32, TB, 0, stream>>>(B0, norm, src, dst, B1, 2.0f, E);

    // gates z (0) and h (2); Wh-side cheb of zeros contributes only bh
    k_mm_wmma<3, false><<<gMM, bMM, 0, stream>>>(x, B0, B1, l1_Wx + 0 * KCC,
                                                 l1_bx + 0 * CH, l1_bh + 0 * CH, B5, 0);
    k_mm_wmma<3, false><<<gMM, bMM, 0, stream>>>(x, B0, B1, l1_Wx + 2 * KCC,
                                                 l1_bx + 2 * CH, l1_bh + 2 * CH, B6, 0);
    k_gate1<<<gNC, TB, 0, stream>>>(B5, B6, bn1_g, bn1_b, bn1_m, bn1_v, B4, NC);

    // ---- layer 2 with hidden state H ----
    // Tx chain from h1
    k_zero<<<gNC, TB, 0, stream>>>(B0, NC);
    k_scatter<<<gE32, TB, 0, stream>>>(B4, norm, src, dst, B0, 1.0f, E);
    k_scalecopy<<<gNC, TB, 0, stream>>>(B1, B4, -1.0f, NC);
    k_scatter<<<gE32, TB, 0, stream>>>(B0, norm, src, dst, B1, 2.0f, E);
    // Th chain from H
    k_zero<<<gNC, TB, 0, stream>>>(B2, NC);
    k_scatter<<<gE32, TB, 0, stream>>>(H, norm, src, dst, B2, 1.0f, E);
    k_scalecopy<<<gNC, TB, 0, stream>>>(B3, H, -1.0f, NC);
    k_scatter<<<gE32, TB, 0, stream>>>(B2, norm, src, dst, B3, 2.0f, E);

    // Z pre-activation
    k_mm_wmma<3, false><<<gMM, bMM, 0, stream>>>(B4, B0, B1, l2_Wx + 0 * KCC,
                                                 l2_bx + 0 * CH, l2_bh + 0 * CH, B5, 0);
    k_mm_wmma<3, false><<<gMM, bMM, 0, stream>>>(H, B2, B3, l2_Wh + 0 * KCC,
                                                 nullptr, nullptr, B5, 1);
    // R pre-activation
    k_mm_wmma<3, false><<<gMM, bMM, 0, stream>>>(B4, B0, B1, l2_Wx + 1 * KCC,
                                                 l2_bx + 1 * CH, l2_bh + 1 * CH, B6, 0);
    k_mm_wmma<3, false><<<gMM, bMM, 0, stream>>>(H, B2, B3, l2_Wh + 1 * KCC,
                                                 nullptr, nullptr, B6, 1);
    // HR = H * sigmoid(R); then its Chebyshev chain (reuse B2/B3)
    k_hr<<<gNC, TB, 0, stream>>>(B6, H, B6, NC);
    k_zero<<<gNC, TB, 0, stream>>>(B2, NC);
    k_scatter<<<gE32, TB, 0, stream>>>(B6, norm, src, dst, B2, 1.0f, E);
    k_scalecopy<<<gNC, TB, 0, stream>>>(B3, B6, -1.0f, NC);
    k_scatter<<<gE32, TB, 0, stream>>>(B2, norm, src, dst, B3, 2.0f, E);

    // Ht pre-activation
    k_mm_wmma<3, false><<<gMM, bMM, 0, stream>>>(B4, B0, B1, l2_Wx + 2 * KCC,
                                                 l2_bx + 2 * CH, l2_bh + 2 * CH, B7, 0);
    k_mm_wmma<3, false><<<gMM, bMM, 0, stream>>>(B6, B2, B3, l2_Wh + 2 * KCC,
                                                 nullptr, nullptr, B7, 1);

    // h = relu(bn2(z*H + (1-z)*tanh(Ht)))   -> second output
    k_gate2<<<gNC, TB, 0, stream>>>(B5, B7, H, bn2_g, bn2_b, bn2_m, bn2_v, hout, NC);

    // out = h @ headW^T + headb   -> first output
    k_mm_wmma<1, true><<<gMM, bMM, 0, stream>>>(hout, nullptr, nullptr, headW,
                                                headb, nullptr, oout, 0);
}